// UnifiedSpectrumCalculator_2259152798237
// MI455X (gfx1250) — compile-verified
//
#include <hip/hip_runtime.h>
#include <math.h>

// ---------------- problem constants (match reference) ----------------
#define NB      400                         // N_BINS
#define NATOMS  1024                        // N_ATOMS
#define NPAIRS  (NATOMS * (NATOMS - 1) / 2) // 523776
#define KW      0.05f                       // KERNEL_WIDTH
#define BSCAT   5.803f                      // b (fm)
#define PI_F    3.14159265358979323846f

typedef __attribute__((ext_vector_type(2))) float v2f;
typedef __attribute__((ext_vector_type(8))) float v8f;

// ---------------------------------------------------------------------
// Small helpers
// ---------------------------------------------------------------------
__device__ __forceinline__ float det3x3(const float* __restrict__ c) {
    // c row-major 3x3
    return c[0] * (c[4] * c[8] - c[5] * c[7])
         - c[1] * (c[3] * c[8] - c[5] * c[6])
         + c[2] * (c[3] * c[7] - c[4] * c[6]);
}

// ---------------------------------------------------------------------
// Kernel 0: zero the KDE accumulator in workspace
// ---------------------------------------------------------------------
__global__ void k_zero(float* __restrict__ s, int n) {
    int i = blockIdx.x * blockDim.x + threadIdx.x;
    if (i < n) s[i] = 0.0f;
}

// ---------------------------------------------------------------------
// Kernel 1: pairwise min-image distances + Gaussian KDE histogram.
// One thread = one upper-triangle pair. All 1024 positions staged in
// LDS (12 KB); per-block LDS histogram (ds_add_f32); one global-atomic
// flush per block. Hot exponential uses the v_exp_f32 TRANS unit, which
// co-executes with VALU on CDNA5.
// ---------------------------------------------------------------------
__global__ void __launch_bounds__(256)
k_kde(const float* __restrict__ pos,
      const float* __restrict__ cell,
      const float* __restrict__ rb_g,
      float* __restrict__ summed) {
    __shared__ float spos[3 * NATOMS];   // 12 KB resident copy of positions
    __shared__ float hist[NB];
    __shared__ float rb[NB];

    for (int t = threadIdx.x; t < 3 * NATOMS; t += blockDim.x) spos[t] = pos[t];
    for (int b = threadIdx.x; b < NB; b += blockDim.x) {
        hist[b] = 0.0f;
        rb[b]   = rb_g[b];
    }
    __syncthreads();

    long long k = (long long)blockIdx.x * blockDim.x + threadIdx.x;
    if (k < (long long)NPAIRS) {
        // ---- decode linear pair index -> (i, j), row-major upper triangle ----
        const double tn = 2.0 * NATOMS - 1.0;
        int i = (int)(0.5 * (tn - sqrt(tn * tn - 8.0 * (double)k)));
        if (i < 0) i = 0;
        if (i > NATOMS - 2) i = NATOMS - 2;
        long long base = (long long)i * (2 * NATOMS - i - 1) / 2;
        while (base > k) { --i; base = (long long)i * (2 * NATOMS - i - 1) / 2; }
        while (base + (NATOMS - 1 - i) <= k) { base += NATOMS - 1 - i; ++i; }
        int j = i + 1 + (int)(k - base);

        // ---- diff = pos[j] - pos[i] (LDS gathers) ----
        float dx = spos[3 * j + 0] - spos[3 * i + 0];
        float dy = spos[3 * j + 1] - spos[3 * i + 1];
        float dz = spos[3 * j + 2] - spos[3 * i + 2];

        // ---- general 3x3 cell inverse (uniform -> scalar regs) ----
        float c00 = cell[0], c01 = cell[1], c02 = cell[2];
        float c10 = cell[3], c11 = cell[4], c12 = cell[5];
        float c20 = cell[6], c21 = cell[7], c22 = cell[8];
        float det = c00 * (c11 * c22 - c12 * c21)
                  - c01 * (c10 * c22 - c12 * c20)
                  + c02 * (c10 * c21 - c11 * c20);
        float id  = 1.0f / det;
        float i00 = (c11 * c22 - c12 * c21) * id, i01 = (c02 * c21 - c01 * c22) * id, i02 = (c01 * c12 - c02 * c11) * id;
        float i10 = (c12 * c20 - c10 * c22) * id, i11 = (c00 * c22 - c02 * c20) * id, i12 = (c02 * c10 - c00 * c12) * id;
        float i20 = (c10 * c21 - c11 * c20) * id, i21 = (c01 * c20 - c00 * c21) * id, i22 = (c00 * c11 - c01 * c10) * id;

        // frac = diff @ cell_inv ; minimum image ; mip = frac @ cell
        float f0 = dx * i00 + dy * i10 + dz * i20;
        float f1 = dx * i01 + dy * i11 + dz * i21;
        float f2 = dx * i02 + dy * i12 + dz * i22;
        f0 -= rintf(f0);  // round-half-even, matches jnp.round
        f1 -= rintf(f1);
        f2 -= rintf(f2);
        float m0 = f0 * c00 + f1 * c10 + f2 * c20;
        float m1 = f0 * c01 + f1 * c11 + f2 * c21;
        float m2 = f0 * c02 + f1 * c12 + f2 * c22;
        float d  = sqrtf(m0 * m0 + m1 * m1 + m2 * m2 + 1e-10f);

        // ---- KDE window + banded accumulation ----
        float rmin = rb[0] - 3.0f * KW;
        float rmax = rb[NB - 1] + 3.0f * KW;
        if (d > rmin && d < rmax) {
            float r0  = rb[0];
            float drb = (rb[NB - 1] - rb[0]) / (float)(NB - 1);
            int   cb  = (int)rintf((d - r0) / drb);
            // +-34 bins ~= +-17 sigma: beyond that exp() underflows f32 to 0,
            // identical to the reference's full 400-bin f32 sum.
            int b0 = cb - 34; if (b0 < 0) b0 = 0;
            int b1 = cb + 34; if (b1 > NB - 1) b1 = NB - 1;
            const float norm = 0.3989422804014327f / KW;  // 1/(w*sqrt(2*pi))
            const float iw   = 1.0f / KW;
            for (int b = b0; b <= b1; ++b) {
                float z = (rb[b] - d) * iw;
                // v_exp_f32 (TRANS): co-executes with VALU; ~1-2 ulp vs ocml
                atomicAdd(&hist[b], norm * __expf(-0.5f * z * z));
            }
        }
    }
    __syncthreads();
    for (int b = threadIdx.x; b < NB; b += blockDim.x) {
        float h = hist[b];
        if (h != 0.0f) atomicAdd(&summed[b], h);
    }
}

// ---------------------------------------------------------------------
// Kernel 2: G(r) normalization -> out[0..NB)
// ---------------------------------------------------------------------
__global__ void k_gofr(const float* __restrict__ cell,
                       const float* __restrict__ rb_g,
                       const float* __restrict__ summed,
                       float* __restrict__ out) {
    int b = blockIdx.x * blockDim.x + threadIdx.x;
    if (b < NB) {
        float vol = fabsf(det3x3(cell));
        float r   = rb_g[b];
        float gp  = (vol / (float)NPAIRS) * summed[b] / (4.0f * PI_F * r * r);
        out[b]    = (BSCAT * BSCAT * 0.01f) * (gp - 1.0f);
    }
}

// ---------------------------------------------------------------------
// Kernel 3: S(Q), F(Q) via f32 WMMA.
// S(q_i) = 1 + 4*pi*rho/(q_i+1e-10) * sum_k sin(q_i r_k) * [W_k r_k G_k]
// -> GEMV with the 400x400 sine matrix. One wave per 16-Q tile; K stepped
// by 4 using V_WMMA_F32_16X16X4_F32 (full f32 precision kept: the sum is
// oscillatory over 400 terms, bf16 would lose ~3 digits).
// B is broadcast across N so every output column equals the GEMV result.
// Full-precision sinf kept here: q*r reaches 250 rad and this stage is
// only ~160k sin calls total.
// ---------------------------------------------------------------------
__global__ void __launch_bounds__(32)
k_sf(const float* __restrict__ cell,
     const float* __restrict__ rb_g,
     const float* __restrict__ qb_g,
     const float* __restrict__ G,
     float* __restrict__ out) {
    __shared__ float gw[NB];   // trapezoid-weight * r * G
    __shared__ float rb[NB];
    __shared__ float qs[16];

    int lane = threadIdx.x;    // 0..31, exactly one wave32 (EXEC all-1s)
    for (int kk = lane; kk < NB; kk += 32) {
        float rm = (kk > 0)      ? rb_g[kk - 1] : rb_g[0];
        float rp = (kk < NB - 1) ? rb_g[kk + 1] : rb_g[NB - 1];
        float Wk = 0.5f * (rp - rm);                // trapezoid weights
        gw[kk]   = Wk * rb_g[kk] * G[kk];
        rb[kk]   = rb_g[kk];
    }
    if (lane < 16) qs[lane] = qb_g[blockIdx.x * 16 + lane];
    __syncthreads();

    // 16x4 f32 A layout (ISA 7.12.2): lane half 0 holds K={0,1} in (x,y),
    // lane half 1 holds K={2,3}; M = lane % 16. B mirrors with M<->N.
    int   M    = lane & 15;
    int   half = lane >> 4;
    int   koff = half * 2;
    float q    = qs[M];

    v8f acc = {};
    for (int kk = 0; kk < NB; kk += 4) {   // 400 % 4 == 0
        int ka = kk + koff;
        v2f a, bv;
        a.x  = sinf(q * rb[ka]);
        a.y  = sinf(q * rb[ka + 1]);
        bv.x = gw[ka];          // broadcast over N: all columns identical
        bv.y = gw[ka + 1];
        acc  = __builtin_amdgcn_wmma_f32_16x16x4_f32(
                   /*neg_a=*/false, a, /*neg_b=*/false, bv,
                   /*c_mod=*/(short)0, acc,
                   /*reuse_a=*/false, /*reuse_b=*/false);
    }

    float vol       = fabsf(det3x3(cell));
    float fourpirho = 4.0f * PI_F * (float)NATOMS / vol;

    // C/D layout: VGPR v, lane half h -> row M = v + 8*h; column N = lane%16.
    // N==0 lanes (lane 0 and lane 16) write the 16 rows of this tile.
    if (M == 0) {
#pragma unroll
        for (int v = 0; v < 8; ++v) {
            int   row = v + half * 8;
            float qv  = qs[row];
            float Sv  = 1.0f + fourpirho * acc[v] / (qv + 1e-10f);
            int   gq  = blockIdx.x * 16 + row;
            out[NB + gq]     = Sv;               // S(Q)
            out[2 * NB + gq] = qv * (Sv - 1.0f); // F(Q)
        }
    }
}

// ---------------------------------------------------------------------
// Host-side launcher (graph-capture safe: only kernel launches on stream)
// ---------------------------------------------------------------------
extern "C" void kernel_launch(void* const* d_in, const int* in_sizes, int n_in,
                              void* d_out, int out_size, void* d_ws, size_t ws_size,
                              hipStream_t stream) {
    const float* positions = (const float*)d_in[0]; // (1024, 3)
    const float* cell      = (const float*)d_in[1]; // (3, 3)
    const float* r_bins    = (const float*)d_in[2]; // (400,)
    const float* q_bins    = (const float*)d_in[3]; // (400,)
    float*       out       = (float*)d_out;         // (3, 400) flat
    float*       summed    = (float*)d_ws;          // NB floats of scratch

    (void)in_sizes; (void)n_in; (void)out_size; (void)ws_size;

    // 0) zero KDE accumulator (deterministic across replays)
    k_zero<<<(NB + 255) / 256, 256, 0, stream>>>(summed, NB);

    // 1) pair distances + Gaussian KDE -> summed[400]
    k_kde<<<(NPAIRS + 255) / 256, 256, 0, stream>>>(positions, cell, r_bins, summed);

    // 2) G(r) -> out[0..400)
    k_gofr<<<(NB + 255) / 256, 256, 0, stream>>>(cell, r_bins, summed, out);

    // 3) S(Q), F(Q) via f32 WMMA sine-transform -> out[400..1200)
    k_sf<<<NB / 16, 32, 0, stream>>>(cell, r_bins, q_bins, out, out);
}